// HiddenMarkovModel_88897233092974
// MI455X (gfx1250) — compile-verified
//
#include <hip/hip_runtime.h>
#include <stdint.h>

// HMM chain simulation, MI455X (gfx1250).
// Store-bound: 400 MB output -> ~17us floor @ 23.3 TB/s.
// 1 thread = 1 particle; 500-step dependent RNG chain in registers; one-hot
// frames staged in LDS (double buffer, 40KB of the 320KB WGP LDS) and drained
// to HBM with CDNA5 async LDS->global B128 stores (ASYNCcnt), overlapping the
// DMA drain with the dependent per-particle RNG chain.

#define NPART   100000   // multiple of 32 -> every wave all-active or all-idle
#define NFRAMES 500
#define FT      10       // frames per tile; 500 = 50 * 10 (even, for B128 pairing)
#define BLOCK   256

typedef int v2i __attribute__((ext_vector_type(2)));
typedef int v4i __attribute__((ext_vector_type(4)));
typedef __attribute__((address_space(1))) v2i as1_v2i;
typedef __attribute__((address_space(3))) v2i as3_v2i;
typedef __attribute__((address_space(1))) v4i as1_v4i;
typedef __attribute__((address_space(3))) v4i as3_v4i;

#if defined(__HIP_DEVICE_COMPILE__)

#define WAIT_DS0() asm volatile("s_wait_dscnt 0" ::: "memory")

#if __has_builtin(__builtin_amdgcn_s_wait_asynccnt)
#define WAIT_ASYNC(n) __builtin_amdgcn_s_wait_asynccnt(n)
#else
#define WAIT_ASYNC(n) asm volatile("s_wait_asynccnt %0" :: "i"(n) : "memory")
#endif

#if __has_builtin(__builtin_amdgcn_global_store_async_from_lds_b128)
#define HAVE_ASYNC_B128 1
#define ASYNC_STORE_B128(g, l)                                         \
  __builtin_amdgcn_global_store_async_from_lds_b128(                   \
      (as1_v4i*)(void*)(g), (as3_v4i*)(void*)(l), 0, 0)
#else
#define HAVE_ASYNC_B128 0
#endif

#if __has_builtin(__builtin_amdgcn_global_store_async_from_lds_b64)
#define ASYNC_STORE_B64(g, l)                                          \
  __builtin_amdgcn_global_store_async_from_lds_b64(                    \
      (as1_v2i*)(void*)(g), (as3_v2i*)(void*)(l), 0, 0)
#else
#define ASYNC_STORE_B64(g, l)                                          \
  asm volatile("global_store_async_from_lds_b64 %0, %1, off"           \
               :: "v"((unsigned long long)(uintptr_t)(g)),             \
                  "v"((unsigned)(uintptr_t)(as3_v2i*)(void*)(l))       \
               : "memory")
#endif

#else  // host pass: bodies are parsed but never codegen'd
#define HAVE_ASYNC_B128       1
#define WAIT_DS0()            ((void)0)
#define WAIT_ASYNC(n)         ((void)0)
#define ASYNC_STORE_B128(g,l) ((void)0)
#define ASYNC_STORE_B64(g, l) ((void)0)
#endif

__device__ __forceinline__ uint64_t splitmix64(uint64_t z) {
  z += 0x9E3779B97F4A7C15ull;
  z = (z ^ (z >> 30)) * 0xBF58476D1CE4E5B9ull;
  z = (z ^ (z >> 27)) * 0x94D049BB133111EBull;
  return z ^ (z >> 31);
}

__global__ __launch_bounds__(BLOCK)
void hmm_chain_kernel(const float* __restrict__ initial,
                      const int* __restrict__ seed_ptr,
                      float* __restrict__ out) {
  // Double-buffered staging: [buf][frame-in-tile][thread] of float2 one-hot.
  __shared__ float2 stage[2][FT * BLOCK];

  const int tid = (int)threadIdx.x;
  const int n   = (int)blockIdx.x * BLOCK + tid;
  const bool active = (n < NPART);   // per-wave all-or-nothing (NPART % 32 == 0)

  const uint32_t seed = (uint32_t)seed_ptr[0];

  // p = U(seed-derived) * 0.001 ; thresholds as u32 compares (no per-step fp).
  const uint64_t zp   = splitmix64((uint64_t)seed ^ 0x243F6A8885A308D3ull);
  const float    up   = (float)(uint32_t)(zp >> 40) * 0x1p-24f;     // [0,1)
  const uint32_t thr1 = (uint32_t)(up * 0.001f * 4294967296.0f);    // p * 2^32
  const uint32_t thr0 = 858993459u;                                 // 0.2 * 2^32

  // Initial state from one-hot input: [1,0] -> 0, [0,1] -> 1.
  uint32_t state = 0u;
  if (active) {
    const float2 ih = ((const float2*)initial)[n];
    state = (ih.y > 0.5f) ? 1u : 0u;
  }

  // Per-particle counter-seeded RNG; LCG64 per step, one uniform per step
  // (the reference also consumes exactly one of its two draws per particle).
  uint64_t rng = splitmix64(((uint64_t)seed << 32) ^
                            ((uint64_t)(uint32_t)n * 0x9E3779B97F4A7C15ull));
  const uint64_t LCG_A = 6364136223846793005ull;
  const uint64_t LCG_C = 1442695040888963407ull;

  // B128 drain mapping: lane L of wave w covers frames 2j+(L>>4) for the
  // particle pair (w*32 + 2*(L&15), +1) -> 16B per lane, 512B per wave per
  // instruction, all sourced from this wave's own LDS rows (s_wait_dscnt is
  // a per-wave fence, so no barrier needed).
  const int lane = tid & 31;
  const int wave = tid >> 5;
  const int half = lane >> 4;                 // which frame of the pair
  const int pcol = wave * 32 + (lane & 15) * 2;  // even particle idx in block
  const int pglb = (int)blockIdx.x * BLOCK + pcol;

  float2* const out2 = (float2*)out;  // [NFRAMES][NPART] of float2

  for (int t = 0; t < NFRAMES / FT; ++t) {
    const int buf = t & 1;

    // Buffer `buf` was handed to the DMA engine 2 tiles ago; at most the
    // previous tile's stores may still be in flight before we overwrite it.
#if HAVE_ASYNC_B128
    WAIT_ASYNC(FT / 2);
#else
    WAIT_ASYNC(FT);
#endif

#pragma unroll
    for (int i = 0; i < FT; ++i) {
      rng = rng * LCG_A + LCG_C;
      const uint32_t u   = (uint32_t)(rng >> 32) ^ (uint32_t)rng;
      const uint32_t thr = state ? thr1 : thr0;
      state ^= (u < thr) ? 1u : 0u;
      const float s1 = (float)state;
      stage[buf][i * BLOCK + tid] = make_float2(1.0f - s1, s1);
    }

    // Make this wave's ds_store results visible before the DMA reads them.
    WAIT_DS0();

    if (active) {
#if HAVE_ASYNC_B128
#pragma unroll
      for (int j = 0; j < FT / 2; ++j) {
        const int fi = 2 * j + half;               // frame-in-tile this lane drains
        const int f  = t * FT + fi;
        float2* gp = out2 + (size_t)f * NPART + pglb;      // 16B, 512B/wave
        float2* lp = &stage[buf][fi * BLOCK + pcol];       // 16B aligned
        ASYNC_STORE_B128(gp, lp);
      }
#else
#pragma unroll
      for (int i = 0; i < FT; ++i) {
        const int f = t * FT + i;
        float2* gp = out2 + (size_t)f * NPART + n;
        float2* lp = &stage[buf][i * BLOCK + tid];
        ASYNC_STORE_B64(gp, lp);
      }
#endif
    }
  }

  // Drain remaining async stores (S_ENDPGM also implies wait-idle).
  WAIT_ASYNC(0);
}

extern "C" void kernel_launch(void* const* d_in, const int* in_sizes, int n_in,
                              void* d_out, int out_size, void* d_ws, size_t ws_size,
                              hipStream_t stream) {
  (void)in_sizes; (void)n_in; (void)out_size; (void)d_ws; (void)ws_size;
  const float* initial  = (const float*)d_in[0];  // [NPART,2] one-hot f32
  const int*   seed_ptr = (const int*)d_in[1];    // scalar
  float*       out      = (float*)d_out;          // [NFRAMES,NPART,2] f32

  dim3 grid((NPART + BLOCK - 1) / BLOCK), block(BLOCK);
  hipLaunchKernelGGL(hmm_chain_kernel, grid, block, 0, stream,
                     initial, seed_ptr, out);
}